// FeatureReconstructionHead_89343909691840
// MI455X (gfx1250) — compile-verified
//
#include <hip/hip_runtime.h>
#include <hip/hip_bf16.h>

// ---------------- problem constants (from reference) ----------------
#define WAY     10
#define RES     49          // 7*7
#define SRDIM   245         // shot*res
#define CDIM    2048
#define HF      32
#define WF      32
#define KROI    40          // n*way
#define NP      1960        // KROI*RES
#define NPPAD   1968        // 123 * 16
#define MT      123         // m tiles of 16
#define NPAD    256         // padded 245 -> 256
#define NCLS4   44
#define WCOLS   100352      // CDIM*RES

typedef __attribute__((ext_vector_type(2))) float v2f;
typedef __attribute__((ext_vector_type(8))) float v8f;
typedef __attribute__((ext_vector_type(4))) int   v4i;

#if defined(__HIP_DEVICE_COMPILE__) && __has_builtin(__builtin_amdgcn_global_load_async_to_lds_b128)
#define HAVE_ASYNC_LDS 1
#endif

// fp32 WMMA 16x16x4:  A is 16x4 (2 VGPRs: lane<16 -> K=0,1 ; lane>=16 -> K=2,3)
//                     B is 4x16 (same striping, N = lane&15)
//                     C/D: row = v + 8*(lane>>4), col = lane&15
__device__ __forceinline__ v8f wmma_f32(v2f a, v2f b, v8f c) {
    return __builtin_amdgcn_wmma_f32_16x16x4_f32(false, a, false, b, (short)0, c,
                                                 false, false);
}

// 16-byte global -> LDS staging (async on CDNA5, plain copy otherwise)
__device__ __forceinline__ void stage16(const float* g, float* l) {
#ifdef HAVE_ASYNC_LDS
    __builtin_amdgcn_global_load_async_to_lds_b128(
        (__attribute__((address_space(1))) v4i*)g,
        (__attribute__((address_space(3))) v4i*)l, 0, 0);
#else
    *(float4*)l = *(const float4*)g;
#endif
}

__device__ __forceinline__ void stage_wait() {
#ifdef HAVE_ASYNC_LDS
#if __has_builtin(__builtin_amdgcn_s_wait_asynccnt)
    __builtin_amdgcn_s_wait_asynccnt(0);
#else
    asm volatile("s_wait_asynccnt 0x0" ::: "memory");
#endif
#endif
}

// ---------------- K0: ROI align -> q (NP x C, row-major) ----------------
__global__ void roi_align_kernel(const float* __restrict__ feat,
                                 const float* __restrict__ rois,
                                 const int*   __restrict__ bidx,
                                 const int*   __restrict__ img_h,
                                 const int*   __restrict__ img_w,
                                 float* __restrict__ q) {
    int kp = blockIdx.x;                 // 0..NP-1
    int k  = kp / RES;
    int p  = kp % RES;
    int oy = p / 7, ox = p % 7;

    float sx = (float)WF / (float)img_w[0];
    float sy = (float)HF / (float)img_h[0];
    float x1 = rois[k * 4 + 0] * sx;
    float y1 = rois[k * 4 + 1] * sy;
    float x2 = rois[k * 4 + 2] * sx;
    float y2 = rois[k * 4 + 3] * sy;
    float bw = fmaxf(x2 - x1, 1.0f) / 7.0f;
    float bh = fmaxf(y2 - y1, 1.0f) / 7.0f;

    int b = bidx[k];

    int   iy0[4], ix0[4], iy1[4], ix1[4];
    float w00[4], w01[4], w10[4], w11[4];
#pragma unroll
    for (int s = 0; s < 4; ++s) {
        int jy = s >> 1, jx = s & 1;
        float yy = y1 + ((float)oy + ((float)jy + 0.5f) * 0.5f) * bh;
        float xx = x1 + ((float)ox + ((float)jx + 0.5f) * 0.5f) * bw;
        yy = fminf(fmaxf(yy, 0.0f), (float)(HF - 1));
        xx = fminf(fmaxf(xx, 0.0f), (float)(WF - 1));
        float fy0 = floorf(yy), fx0 = floorf(xx);
        int y0 = (int)fy0, x0 = (int)fx0;
        int y1i = min(y0 + 1, HF - 1), x1i = min(x0 + 1, WF - 1);
        float wy = yy - fy0, wx = xx - fx0;
        iy0[s] = y0; ix0[s] = x0; iy1[s] = y1i; ix1[s] = x1i;
        w00[s] = (1.0f - wy) * (1.0f - wx);
        w01[s] = (1.0f - wy) * wx;
        w10[s] = wy * (1.0f - wx);
        w11[s] = wy * wx;
    }

    for (int c = threadIdx.x; c < CDIM; c += blockDim.x) {
        const float* fc = feat + ((size_t)b * CDIM + c) * (HF * WF);
        float sum = 0.0f;
#pragma unroll
        for (int s = 0; s < 4; ++s) {
            float f00 = fc[iy0[s] * WF + ix0[s]];
            float f01 = fc[iy0[s] * WF + ix1[s]];
            float f10 = fc[iy1[s] * WF + ix0[s]];
            float f11 = fc[iy1[s] * WF + ix1[s]];
            sum += f00 * w00[s] + f01 * w01[s] + f10 * w10[s] + f11 * w11[s];
        }
        q[(size_t)kp * CDIM + c] = sum * 0.25f;
    }
}

// generic 16x16 WMMA row-tile x row-tile^T inner product over K=CDIM.
// Masks are hoisted: rows are clamped, fragments scaled by 0/1 masks.
__device__ __forceinline__ v8f gemm_rowrow(const float* __restrict__ baseA, int m0,
                                           int maxA,
                                           const float* __restrict__ baseB, int n0,
                                           int maxB, int half, int rr) {
    int rowA = m0 + rr, rowB = n0 + rr;
    float ma = (rowA < maxA) ? 1.0f : 0.0f;
    float mb = (rowB < maxB) ? 1.0f : 0.0f;
    const float* pa = baseA + (size_t)min(rowA, maxA - 1) * CDIM + 2 * half;
    const float* pb = baseB + (size_t)min(rowB, maxB - 1) * CDIM + 2 * half;
    __builtin_prefetch(pa, 0, 3);
    __builtin_prefetch(pb, 0, 3);

    v8f acc = {};
    if ((m0 + 15 < maxA) && (n0 + 15 < maxB)) {   // interior: no masks at all
        for (int k = 0; k < CDIM; k += 4) {
            v2f a = {pa[k], pa[k + 1]};
            v2f b = {pb[k], pb[k + 1]};
            acc = wmma_f32(a, b, acc);
        }
    } else {
        for (int k = 0; k < CDIM; k += 4) {
            v2f a = {pa[k], pa[k + 1]};
            v2f b = {pb[k], pb[k + 1]};
            a *= ma;
            b *= mb;
            acc = wmma_f32(a, b, acc);
        }
    }
    return acc;
}

// ------- K1: A_w = S_w S_w^T + lam*I, padded to 256x256 (pad block = I) -------
__global__ void gram_kernel(const float* __restrict__ S,
                            const float* __restrict__ rvec,
                            float* __restrict__ A) {
    int wid = threadIdx.x >> 5;
    int tn = blockIdx.x * 4 + wid, tm = blockIdx.y, w = blockIdx.z;
    const float* Sw = S + (size_t)w * SRDIM * CDIM;
    int lane = threadIdx.x & 31;
    int half = lane >> 4, rr = lane & 15;
    int m0 = tm * 16, n0 = tn * 16;

    v8f acc = gemm_rowrow(Sw, m0, SRDIM, Sw, n0, SRDIM, half, rr);

    float lam = ((float)SRDIM / (float)CDIM) * expf(rvec[0]) + 1e-6f;
#pragma unroll
    for (int v = 0; v < 8; ++v) {
        int row = m0 + v + 8 * half;
        int col = n0 + rr;
        float val = acc[v];
        if (row == col) val += (row < SRDIM) ? lam : 1.0f;  // pad diag -> 1
        A[((size_t)w * NPAD + row) * NPAD + col] = val;
    }
}

// ---------- K2: Gauss-Jordan inverse of 256x256 (one block per way) ----------
__global__ void gj_inverse_kernel(float* __restrict__ A, float* __restrict__ Inv) {
    int w = blockIdx.x;
    A   += (size_t)w * NPAD * NPAD;
    Inv += (size_t)w * NPAD * NPAD;
    int t = threadIdx.x;  // 256 threads, thread t owns column t

    for (int i = 0; i < NPAD; ++i) Inv[i * NPAD + t] = (i == t) ? 1.0f : 0.0f;
    __syncthreads();

    __shared__ float pr[NPAD];   // scaled pivot row of A
    __shared__ float pi[NPAD];   // scaled pivot row of Inv
    __shared__ float col[NPAD];  // pivot column snapshot

    for (int piv = 0; piv < NPAD; ++piv) {
        float invd = 1.0f / A[piv * NPAD + piv];
        pr[t]  = A[piv * NPAD + t] * invd;
        pi[t]  = Inv[piv * NPAD + t] * invd;
        col[t] = A[t * NPAD + piv];
        __syncthreads();
        for (int i = 0; i < NPAD; ++i) {
            if (i == piv) {
                A[i * NPAD + t]   = pr[t];
                Inv[i * NPAD + t] = pi[t];
            } else {
                float f = col[i];
                A[i * NPAD + t]   -= f * pr[t];
                Inv[i * NPAD + t] -= f * pi[t];
            }
        }
        __syncthreads();
    }
}

// ---------------- K3: T_w = q @ S_w^T   (NPPAD x 256 per way) ----------------
__global__ void gemm_T_kernel(const float* __restrict__ q,
                              const float* __restrict__ S,
                              float* __restrict__ T) {
    int wid = threadIdx.x >> 5;
    int tn = blockIdx.x * 4 + wid, tm = blockIdx.y, w = blockIdx.z;
    const float* Sw = S + (size_t)w * SRDIM * CDIM;
    int lane = threadIdx.x & 31;
    int half = lane >> 4, rr = lane & 15;
    int m0 = tm * 16, n0 = tn * 16;

    v8f acc = gemm_rowrow(q, m0, NP, Sw, n0, SRDIM, half, rr);

#pragma unroll
    for (int v = 0; v < 8; ++v) {
        int row = m0 + v + 8 * half;
        int col = n0 + rr;
        T[((size_t)w * NPPAD + row) * NPAD + col] = acc[v];
    }
}

// ---------------- K4: V_w = T_w @ Ainv_w  (NPPAD x 256) ----------------
__global__ void gemm_V_kernel(const float* __restrict__ T,
                              const float* __restrict__ Ainv,
                              float* __restrict__ V) {
    int wid = threadIdx.x >> 5;
    int tn = blockIdx.x * 4 + wid, tm = blockIdx.y, w = blockIdx.z;
    const float* Tw = T    + (size_t)w * NPPAD * NPAD;
    const float* Aw = Ainv + (size_t)w * NPAD * NPAD;
    int lane = threadIdx.x & 31;
    int half = lane >> 4, rr = lane & 15;
    int m0 = tm * 16, n0 = tn * 16;

    const float* pa = Tw + (size_t)(m0 + rr) * NPAD + 2 * half;
    const float* pb = Aw + (size_t)(2 * half) * NPAD + n0 + rr;
    __builtin_prefetch(pa, 0, 3);
    __builtin_prefetch(pb, 0, 3);

    v8f acc = {};
    for (int k = 0; k < NPAD; k += 4) {
        v2f a = {pa[k], pa[k + 1]};
        v2f b = {pb[(size_t)k * NPAD], pb[(size_t)(k + 1) * NPAD]};
        acc = wmma_f32(a, b, acc);
    }
#pragma unroll
    for (int v = 0; v < 8; ++v) {
        int row = m0 + v + 8 * half;
        int col = n0 + rr;
        V[((size_t)w * NPPAD + row) * NPAD + col] = acc[v];
    }
}

// -------- K5: fused Z = rho * V_w @ S_w ; eu[w][m] = sum_c (Z - q)^2 --------
// 4 waves/block share the LDS V-tile; each wave covers 1/4 of the C chunks.
__global__ void fused_eu_kernel(const float* __restrict__ V,
                                const float* __restrict__ S,
                                const float* __restrict__ q,
                                const float* __restrict__ rvec,
                                float* __restrict__ eu) {
    int tm = blockIdx.x, w = blockIdx.y;
    const float* Vw = V + (size_t)w * NPPAD * NPAD;
    const float* Sw = S + (size_t)w * SRDIM * CDIM;
    int wid  = threadIdx.x >> 5;
    int lane = threadIdx.x & 31;
    int half = lane >> 4, rr = lane & 15;
    int m0 = tm * 16;
    float rho = expf(rvec[1]);

    __shared__ __align__(16) float vt[16 * NPAD];  // 16 KB V tile (contiguous rows)
    __shared__ float esum[16];
    const float* vsrc = Vw + (size_t)m0 * NPAD;
    for (int i = threadIdx.x; i < 16 * NPAD / 4; i += blockDim.x)
        stage16(vsrc + i * 4, vt + i * 4);
    if (threadIdx.x < 16) esum[threadIdx.x] = 0.0f;
    stage_wait();
    __syncthreads();

    // hoisted q-row masks (rows >= NP have V==0 -> z==0, so masked q is exact)
    const float* pq[8];
    float mq[8];
#pragma unroll
    for (int v = 0; v < 8; ++v) {
        int gm = m0 + v + 8 * half;
        mq[v] = (gm < NP) ? 1.0f : 0.0f;
        pq[v] = q + (size_t)min(gm, NP - 1) * CDIM;
    }
    // K-tail fragment setup (chunk k=244 covers cols 244..247; only 244 valid)
    int kr0 = 244 + 2 * half;
    float mb0 = (kr0 < SRDIM) ? 1.0f : 0.0f;
    const float* ptail = Sw + (size_t)min(kr0, SRDIM - 1) * CDIM;
    const float* vtrow = vt + rr * NPAD;

    float eacc[8];
#pragma unroll
    for (int v = 0; v < 8; ++v) eacc[v] = 0.0f;

    for (int c0 = wid * 16; c0 < CDIM; c0 += 64) {
        const float* pb = Sw + (size_t)(2 * half) * CDIM + c0 + rr;
        v8f acc = {};
        for (int k = 0; k < 244; k += 4) {        // cols 0..243: unguarded
            v2f a = {vtrow[k + 2 * half], vtrow[k + 2 * half + 1]};
            v2f b = {pb[(size_t)k * CDIM], pb[(size_t)(k + 1) * CDIM]};
            acc = wmma_f32(a, b, acc);
        }
        {   // masked tail: col 244 (cols 245..255 contribute exactly zero)
            v2f a = {vtrow[kr0], vtrow[kr0 + 1]};
            v2f b = {ptail[c0 + rr] * mb0, 0.0f};
            acc = wmma_f32(a, b, acc);
        }
#pragma unroll
        for (int v = 0; v < 8; ++v) {
            float z  = rho * acc[v];
            float qv = pq[v][c0 + rr] * mq[v];
            float d  = z - qv;
            eacc[v] += d * d;
        }
    }

    // reduce over the 16 lanes of each half (columns), combine waves via LDS
#pragma unroll
    for (int v = 0; v < 8; ++v) {
        float s = eacc[v];
        for (int m = 1; m < 16; m <<= 1) s += __shfl_xor(s, m, 32);
        if (rr == 0) atomicAdd(&esum[v + 8 * half], s);
    }
    __syncthreads();
    if (threadIdx.x < 16)
        eu[(size_t)w * NPPAD + m0 + threadIdx.x] = esum[threadIdx.x];
}

// -------- K6: metric + log_softmax -> d_out[1760 .. 2160) --------
__global__ void metric_kernel(const float* __restrict__ eu,
                              const float* __restrict__ scale,
                              float* __restrict__ out) {
    int k = blockIdx.x;          // 0..39
    int t = threadIdx.x;         // 32 threads
    __shared__ float lg[WAY];
    if (t < WAY) {
        float s = 0.0f;
        for (int p = 0; p < RES; ++p) s += eu[(size_t)t * NPPAD + k * RES + p];
        lg[t] = -(s / (float)RES) * scale[0];
    }
    __syncthreads();
    if (t < WAY) {
        float mx = -3.402823e38f;
        for (int wv = 0; wv < WAY; ++wv) mx = fmaxf(mx, lg[wv]);
        float se = 0.0f;
        for (int wv = 0; wv < WAY; ++wv) se += expf(lg[wv] - mx);
        out[KROI * NCLS4 + k * WAY + t] = lg[t] - mx - logf(se);
    }
}

// -------- K7: roi_cls_locs = rf_flat @ W_cls^T + b -> d_out[0 .. 1760) --------
__global__ void cls_head_kernel(const float* __restrict__ q,
                                const float* __restrict__ Wc,
                                const float* __restrict__ bc,
                                float* __restrict__ out) {
    int o = blockIdx.x;   // 0..43
    int k = blockIdx.y;   // 0..39
    __shared__ float red[256];
    float acc = 0.0f;
    const float* wrow = Wc + (size_t)o * WCOLS;
    for (int idx = threadIdx.x; idx < WCOLS; idx += 256) {
        int c = idx / RES;
        int p = idx % RES;   // col layout of rf.reshape(K,-1): c*49 + p
        acc += wrow[idx] * q[((size_t)k * RES + p) * CDIM + c];
    }
    red[threadIdx.x] = acc;
    __syncthreads();
    for (int s = 128; s > 0; s >>= 1) {
        if (threadIdx.x < s) red[threadIdx.x] += red[threadIdx.x + s];
        __syncthreads();
    }
    if (threadIdx.x == 0) out[k * NCLS4 + o] = red[0] + bc[o];
}

// ---------------------------------------------------------------------------
extern "C" void kernel_launch(void* const* d_in, const int* in_sizes, int n_in,
                              void* d_out, int out_size, void* d_ws, size_t ws_size,
                              hipStream_t stream) {
    const float* support = (const float*)d_in[0];   // (10, 245, 2048)
    const float* query   = (const float*)d_in[1];   // (4, 2048, 32, 32)
    const float* rois    = (const float*)d_in[2];   // (40, 4)
    const int*   ridx    = (const int*)d_in[3];     // (40,)
    const float* rvec    = (const float*)d_in[4];   // (2,)
    const float* scale   = (const float*)d_in[5];   // (1,)
    const float* Wc      = (const float*)d_in[6];   // (44, 100352)
    const float* bc      = (const float*)d_in[7];   // (44,)
    const int*   img_h   = (const int*)d_in[8];
    const int*   img_w   = (const int*)d_in[9];
    float* out = (float*)d_out;

    float* ws = (float*)d_ws;
    size_t off = 0;
    float* q    = ws + off; off += (size_t)NP * CDIM;
    float* A    = ws + off; off += (size_t)WAY * NPAD * NPAD;
    float* Ainv = ws + off; off += (size_t)WAY * NPAD * NPAD;
    float* T    = ws + off; off += (size_t)WAY * NPPAD * NPAD;
    float* V    = ws + off; off += (size_t)WAY * NPPAD * NPAD;
    float* eu   = ws + off; off += (size_t)WAY * NPPAD;

    // K0: ROI-align -> q
    roi_align_kernel<<<dim3(NP), dim3(256), 0, stream>>>(query, rois, ridx,
                                                         img_h, img_w, q);
    // K1: A = S S^T + lam I (padded)
    gram_kernel<<<dim3(4, 16, WAY), dim3(128), 0, stream>>>(support, rvec, A);
    // K2: Ainv = A^-1 (Gauss-Jordan, PD so no pivoting)
    gj_inverse_kernel<<<dim3(WAY), dim3(NPAD), 0, stream>>>(A, Ainv);
    // K3: T = q S^T
    gemm_T_kernel<<<dim3(4, MT, WAY), dim3(128), 0, stream>>>(q, support, T);
    // K4: V = T Ainv
    gemm_V_kernel<<<dim3(4, MT, WAY), dim3(128), 0, stream>>>(T, Ainv, V);
    // K5: eu = rowsum((rho * V S - q)^2)
    fused_eu_kernel<<<dim3(MT, WAY), dim3(128), 0, stream>>>(V, support, q, rvec, eu);
    // K6: metric + log_softmax
    metric_kernel<<<dim3(KROI), dim3(32), 0, stream>>>(eu, scale, out);
    // K7: classification head
    cls_head_kernel<<<dim3(NCLS4, KROI), dim3(256), 0, stream>>>(q, Wc, bc, out);
}